// DNAEmbedding_5111011082276
// MI455X (gfx1250) — compile-verified
//
#include <hip/hip_runtime.h>

typedef __attribute__((ext_vector_type(2))) float v2f;
typedef __attribute__((ext_vector_type(4))) float v4f;
typedef __attribute__((ext_vector_type(8))) float v8f;

#define HIDDEN 768
#define DDIM   192
#define CDIM   960          // HIDDEN + DDIM
#define NCOMBO 136          // 128 (tok x dinuc) + 8 (last-position, zero dinuc)
#define MPAD   144          // 9 * 16 WMMA M-tiles
#define BATCH  32
#define SEQ    2048
#define NTOK   (BATCH * SEQ)
#define LN_EPS 1e-12f

// ---------------------------------------------------------------------------
// Kernel 1: build the 144x960 concat table (rows >= 136 zero-padded).
//   c <  128 : row = [token_emb[c>>4] | dinuc_emb[c&15]]
//   c >= 128 : row = [token_emb[c-128] | zeros]   (padded last seq position)
// ---------------------------------------------------------------------------
__global__ void build_ccat_kernel(const float* __restrict__ tok_emb,
                                  const float* __restrict__ din_emb,
                                  float* __restrict__ ccat) {
    int c = blockIdx.x;
    float* row = ccat + c * CDIM;
    if (c >= NCOMBO) {
        for (int i = threadIdx.x; i < CDIM; i += blockDim.x) row[i] = 0.0f;
        return;
    }
    int t, d;
    bool has_d;
    if (c < 128) { t = c >> 4;  d = c & 15; has_d = true;  }
    else         { t = c - 128; d = 0;      has_d = false; }
    for (int i = threadIdx.x; i < CDIM; i += blockDim.x) {
        float v;
        if (i < HIDDEN) v = tok_emb[t * HIDDEN + i];
        else            v = has_d ? din_emb[d * DDIM + (i - HIDDEN)] : 0.0f;
        row[i] = v;
    }
}

// ---------------------------------------------------------------------------
// Kernel 2: table GEMM  X[m, n] = sum_k ccat[m, k] * W[n, k]
// using v_wmma_f32_16x16x4_f32 tiles (fp32 precision, matches reference).
// One wave (32 lanes) per 16x16 output tile. grid = (48 N-tiles, 9 M-tiles).
// Layouts per CDNA5 ISA 7.12.2:
//   A 16x4 : lanes 0-15 M=0..15 {K=0,K=1}; lanes 16-31 M=0..15 {K=2,K=3}
//   B 4x16 : lanes 0-15 N=0..15 {K=0,K=1}; lanes 16-31 N=0..15 {K=2,K=3}
//   C/D    : vgpr v, lane half h -> M = v + 8h, N = lane&15
// ---------------------------------------------------------------------------
__global__ void table_gemm_wmma(const float* __restrict__ ccat,
                                const float* __restrict__ W,
                                float* __restrict__ xtab) {
    const int nbase  = blockIdx.x * 16;
    const int mbase  = blockIdx.y * 16;
    const int l      = threadIdx.x;      // 0..31, EXEC all ones
    const int half   = l >> 4;
    const int lane16 = l & 15;

    const float* arow = ccat + (mbase + lane16) * CDIM;
    const float* brow = W    + (nbase + lane16) * CDIM;

    v8f acc = {};
    for (int k0 = 0; k0 < CDIM; k0 += 4) {
        const int kk = k0 + 2 * half;
        v2f a, b;
        a.x = arow[kk]; a.y = arow[kk + 1];
        b.x = brow[kk]; b.y = brow[kk + 1];
        acc = __builtin_amdgcn_wmma_f32_16x16x4_f32(
            /*neg_a=*/false, a, /*neg_b=*/false, b,
            /*c_mod=*/(short)0, acc, /*reuse_a=*/false, /*reuse_b=*/false);
    }

    #pragma unroll
    for (int v = 0; v < 8; ++v) {
        const int m = mbase + v + 8 * half;      // m < MPAD always
        xtab[m * HIDDEN + nbase + lane16] = acc[v];
    }
}

// ---------------------------------------------------------------------------
// Kernel 3: per-row bias + LayerNorm over the 136 valid table rows.
// 256 threads / row, 3 elements per thread (768 = 3*256). Two-pass mean/var.
// ---------------------------------------------------------------------------
__global__ void ln_rows_kernel(const float* __restrict__ xtab,
                               const float* __restrict__ bias,
                               const float* __restrict__ gamma,
                               const float* __restrict__ beta,
                               float* __restrict__ ytab) {
    const int row = blockIdx.x;
    const int tid = threadIdx.x;
    const float* xr = xtab + row * HIDDEN;

    float x0 = xr[tid]       + bias[tid];
    float x1 = xr[tid + 256] + bias[tid + 256];
    float x2 = xr[tid + 512] + bias[tid + 512];

    __shared__ float wred[8];
    __shared__ float bcast;
    const int wid = tid >> 5, lane = tid & 31;

    // mean
    float s = x0 + x1 + x2;
    for (int o = 16; o > 0; o >>= 1) s += __shfl_down(s, o, 32);
    if (lane == 0) wred[wid] = s;
    __syncthreads();
    if (tid == 0) {
        float tot = 0.0f;
        for (int i = 0; i < 8; ++i) tot += wred[i];
        bcast = tot * (1.0f / (float)HIDDEN);
    }
    __syncthreads();
    const float mu = bcast;

    // variance
    const float d0 = x0 - mu, d1 = x1 - mu, d2 = x2 - mu;
    float v = d0 * d0 + d1 * d1 + d2 * d2;
    for (int o = 16; o > 0; o >>= 1) v += __shfl_down(v, o, 32);
    if (lane == 0) wred[wid] = v;
    __syncthreads();
    if (tid == 0) {
        float tot = 0.0f;
        for (int i = 0; i < 8; ++i) tot += wred[i];
        bcast = 1.0f / sqrtf(tot * (1.0f / (float)HIDDEN) + LN_EPS);
    }
    __syncthreads();
    const float inv = bcast;

    float* yr = ytab + row * HIDDEN;
    yr[tid]       = d0 * inv * gamma[tid]       + beta[tid];
    yr[tid + 256] = d1 * inv * gamma[tid + 256] + beta[tid + 256];
    yr[tid + 512] = d2 * inv * gamma[tid + 512] + beta[tid + 512];
}

// ---------------------------------------------------------------------------
// Kernel 4: gather — the only HBM-heavy phase (~192 MB streamed out).
// One block per token: resolve combo index, stream one 768-float table row
// (L2 resident) to d_out with non-temporal 16B stores. 192 thr = 192 v4f.
// ---------------------------------------------------------------------------
__global__ void gather_rows_kernel(const int* __restrict__ ids,
                                   const float* __restrict__ ytab,
                                   float* __restrict__ out) {
    const int tok = blockIdx.x;
    const int s   = tok & (SEQ - 1);
    const int t   = ids[tok];

    int c;
    if (s == SEQ - 1) {
        c = 128 + t;                               // padded last position
    } else {
        const int nx = ids[tok + 1];
        const int d  = ((t >= 4) && (nx >= 4)) ? ((t - 4) * 4 + (nx - 4)) : 0;
        c = t * 16 + d;
    }

    const v4f* src = (const v4f*)(ytab + c * HIDDEN);
    v4f*       dst = (v4f*)(out + (size_t)tok * HIDDEN);
    __builtin_nontemporal_store(src[threadIdx.x], &dst[threadIdx.x]);
}

// ---------------------------------------------------------------------------
extern "C" void kernel_launch(void* const* d_in, const int* in_sizes, int n_in,
                              void* d_out, int out_size, void* d_ws, size_t ws_size,
                              hipStream_t stream) {
    const int*   ids     = (const int*)  d_in[0];   // input_ids
    const float* tok_emb = (const float*)d_in[1];   // token_emb  (8 x 768)
    const float* din_emb = (const float*)d_in[2];   // dinuc_emb  (16 x 192)
    const float* proj_W  = (const float*)d_in[3];   // proj_W     (768 x 960)
    const float* proj_b  = (const float*)d_in[4];   // proj_b     (768)
    const float* gamma   = (const float*)d_in[5];   // ln_gamma   (768)
    const float* beta    = (const float*)d_in[6];   // ln_beta    (768)
    float*       out     = (float*)d_out;

    // workspace layout (all freshly written each call; total < 2 MB)
    char* ws   = (char*)d_ws;
    float* ccat = (float*)(ws);                         // 144*960*4  = 552960 B
    float* xtab = (float*)(ws + 576 * 1024);            // 144*768*4  = 442368 B
    float* ytab = (float*)(ws + 1088 * 1024);           // 136*768*4  = 417792 B

    build_ccat_kernel<<<MPAD, 256, 0, stream>>>(tok_emb, din_emb, ccat);
    table_gemm_wmma<<<dim3(HIDDEN / 16, MPAD / 16), 32, 0, stream>>>(ccat, proj_W, xtab);
    ln_rows_kernel<<<NCOMBO, 256, 0, stream>>>(xtab, proj_b, gamma, beta, ytab);
    gather_rows_kernel<<<NTOK, HIDDEN / 4, 0, stream>>>(ids, ytab, out);
}